// WaveletSparsityPrior_31301721653341
// MI455X (gfx1250) — compile-verified
//
#include <hip/hip_runtime.h>

// ============================================================================
// WaveletSparsityPrior on MI455X (gfx1250)
//
// pred: (64,1,1024,1024) f32 -> scalar loss. Pure streaming reduction:
// 256 MB / 23.3 TB/s ~= 11 us floor. The 3-level Haar DWT is tile-local to an
// aligned 16x16 tile and is computed as matrix sandwiches on the f32 matrix
// pipe (V_WMMA_F32_16X16X4_F32):  D = M*(M*Z)^T == (M Z M^T)^T, two
// left-multiplies with a conflict-free LDS transpose between them.
// Transposed-ness alternates per level; band sets are transpose-symmetric.
// Haar matrices use exact +-1 entries; 1/sqrt(2) factors are folded into
// per-level thresholds (thr * 2^j) and final scales (w_j / (3 * N_j * 2^j)).
//
// Data movement: tile staged into per-wave LDS via async global->LDS b128
// copies (ASYNCcnt path) when the builtin exists; direct-load fallback kept.
// ============================================================================

typedef float v2f __attribute__((ext_vector_type(2)));
typedef float v8f __attribute__((ext_vector_type(8)));

#define IMG      1024
#define TILES_X  64                    // 1024/16
#define PITCH    20                    // LDS row pitch (floats): conflict-free

#define BASE_THR (50.0f / 255.0f)
// unscaled coeff at level j = 2^j * true coeff  =>  clip at 2^j * thr_true
#define THR1 (BASE_THR * 0.5f)         // 2^1 * (BASE/4)
#define THR2 (BASE_THR * 2.0f)         // 2^2 * (BASE/2)
#define THR3 (BASE_THR * 8.0f)         // 2^3 *  BASE
// scale_j = weight_j / (3 * N_j * 2^j);  N_j = 64*(1024>>j)^2
#define S1 (1.0f / 301989888.0f)       // (1/3)/(3*16777216*2)
#define S2 (1.0f / 100663296.0f)       // (1/2)/(3*4194304*4)
#define S3 (1.0f /  25165824.0f)       //   1  /(3*1048576*8)

#if __has_builtin(__builtin_amdgcn_global_load_async_to_lds_b128)
#define HAVE_ASYNC_LDS 1
#define AS_GLOBAL __attribute__((address_space(1)))
#define AS_LDS    __attribute__((address_space(3)))
// exact pointee type from the compiler diagnostic: int __vector_size__(16)
typedef int b128_t __attribute__((vector_size(4 * sizeof(int))));
#endif

// Entry (i,c) of the level operator: Haar(+-1) of size nlev embedded in the
// top-left of a 16x16 identity. Branchless (v_cndmask), no EXEC divergence.
__device__ __forceinline__ float op_entry(int i, int c, int nlev) {
    const int half = nlev >> 1;
    float lo = ((i < half) && (c == 2 * i || c == 2 * i + 1)) ? 1.0f : 0.0f;
    const int p = 2 * (i - half);
    float hv = (c == p) ? -1.0f : ((c == p + 1) ? 1.0f : 0.0f);
    float hi = ((i >= half) && (i < nlev)) ? hv : 0.0f;
    float id = ((i >= nlev) && (c == i)) ? 1.0f : 0.0f;
    return lo + hi + id;
}

// D(16x16) = A * B, via 4 chained K=4 f32 WMMAs (A slices precomputed).
__device__ __forceinline__ v8f haar_mul(const v2f* A, const v2f* B) {
    v8f d = 0.0f;
#pragma unroll
    for (int k = 0; k < 4; ++k) {
        d = __builtin_amdgcn_wmma_f32_16x16x4_f32(
                /*neg_a=*/false, A[k], /*neg_b=*/false, B[k],
                /*c_mod=*/(short)0, d, /*reuse_a=*/false, /*reuse_b=*/false);
    }
    return d;
}

// D layout: lane l, vgpr g -> (M = g + 8*(l>=16), N = l%16).
__device__ __forceinline__ void store_tile(float* T, const v8f& d, int n, int h) {
#pragma unroll
    for (int g = 0; g < 8; ++g) T[(g + 8 * h) * PITCH + n] = d[g];
}
// B layout (4x16 f32, 2 VGPRs): lane -> col N = lane%16; K = vgpr + 2*(lane>=16).
__device__ __forceinline__ void load_B(const float* T, v2f* B, int n, int h, bool transp) {
#pragma unroll
    for (int k = 0; k < 4; ++k) {
        const int r0 = 4 * k + 0 + 2 * h;
        const int r1 = 4 * k + 1 + 2 * h;
        if (transp) { B[k].x = T[n * PITCH + r0]; B[k].y = T[n * PITCH + r1]; }
        else        { B[k].x = T[r0 * PITCH + n]; B[k].y = T[r1 * PITCH + n]; }
    }
}

extern "C" __global__ void __launch_bounds__(256)
haar_wavelet_loss(const float* __restrict__ pred, float* __restrict__ partials) {
    __shared__ float lds[8 * 16 * PITCH];   // per-wave private 16x16 (pitched)
    __shared__ float wsum[8];

    const int lane = threadIdx.x & 31;
    const int wid  = threadIdx.x >> 5;
    const int n    = lane & 15;             // column / row-within-half
    const int h    = lane >> 4;             // lane half
    float* T = lds + wid * (16 * PITCH);

    const int gwave = blockIdx.x * 8 + wid; // one wave per 16x16 tile
    const int tx = gwave & (TILES_X - 1);
    const int ty = (gwave / TILES_X) & (TILES_X - 1);
    const int b  = gwave / (TILES_X * TILES_X);

    const float* base = pred + (size_t)b * (IMG * IMG)
                             + (size_t)(ty * 16) * IMG + (size_t)(tx * 16);

    v2f B[4];

#ifdef HAVE_ASYNC_LDS
    // ---- stage tile into LDS on the async (ASYNCcnt) pipe: 2 x b128/wave ----
    {
        const int r   = lane >> 2;               // row 0..7
        const int seg = (lane & 3) * 4;          // 4-float segment in row
        float* g0 = (float*)(base + (size_t)r * IMG + seg);
        float* g1 = (float*)(base + (size_t)(r + 8) * IMG + seg);
        float* l0 = T + r * PITCH + seg;
        float* l1 = T + (r + 8) * PITCH + seg;
        __builtin_amdgcn_global_load_async_to_lds_b128(
            (AS_GLOBAL b128_t*)g0, (AS_LDS b128_t*)l0, 0, 0);
        __builtin_amdgcn_global_load_async_to_lds_b128(
            (AS_GLOBAL b128_t*)g1, (AS_LDS b128_t*)l1, 0, 0);
    }
#endif

    // ---- precompute the three level operators' A-slices (used twice each) --
    v2f A16[4], A8[4], A4[4];
#pragma unroll
    for (int k = 0; k < 4; ++k) {
        const int c0 = 4 * k + 0 + 2 * h, c1 = 4 * k + 1 + 2 * h;
        A16[k].x = op_entry(n, c0, 16);  A16[k].y = op_entry(n, c1, 16);
        A8[k].x  = op_entry(n, c0,  8);  A8[k].y  = op_entry(n, c1,  8);
        A4[k].x  = op_entry(n, c0,  4);  A4[k].y  = op_entry(n, c1,  4);
    }

#ifdef HAVE_ASYNC_LDS
#if __has_builtin(__builtin_amdgcn_s_wait_asynccnt)
    __builtin_amdgcn_s_wait_asynccnt(0);
#else
    asm volatile("s_wait_asynccnt 0x0" ::: "memory");
#endif
    asm volatile("" ::: "memory");
    load_B(T, B, n, h, /*transp=*/false);   // X slices from LDS
#else
    // ---- fallback: direct per-lane loads into B-operand layout ------------
#pragma unroll
    for (int k = 0; k < 4; ++k) {
        B[k].x = base[(size_t)(4 * k + 0 + 2 * h) * IMG + n];
        B[k].y = base[(size_t)(4 * k + 1 + 2 * h) * IMG + n];
    }
#endif

    v8f d;
    float total = 0.0f;

    // ---------------- level 1 : full 16x16 sandwich (H16) ----------------
    d = haar_mul(A16, B);                   // D1 = H*X
    store_tile(T, d, n, h);
    load_B(T, B, n, h, /*transp=*/true);    // B = D1^T slices
    d = haar_mul(A16, B);                   // D2 = H*(H*X)^T = Y^T
    {
        float m1 = (h == 0 && n < 8) ? 0.0f : 1.0f;  // exclude LL quadrant
        float acc = 0.0f;
#pragma unroll
        for (int g = 0; g < 8; ++g) acc += fminf(fabsf(d[g]), THR1);
        total += S1 * m1 * acc;
    }

    // ---------------- level 2 : H8 (+identity padding) --------------------
    store_tile(T, d, n, h);
    load_B(T, B, n, h, /*transp=*/false);
    d = haar_mul(A8, B);                    // G*D2
    store_tile(T, d, n, h);
    load_B(T, B, n, h, /*transp=*/true);
    d = haar_mul(A8, B);                    // D4 = G*Y*G^T  (untransposed)
    {
        float mA = (h == 0 && n >= 4 && n < 8) ? 1.0f : 0.0f; // rows M<4
        float mB = (h == 0 && n < 8) ? 1.0f : 0.0f;           // rows M=4..7
        float accA = 0.0f, accB = 0.0f;
#pragma unroll
        for (int g = 0; g < 4; ++g) accA += fminf(fabsf(d[g]), THR2);
#pragma unroll
        for (int g = 4; g < 8; ++g) accB += fminf(fabsf(d[g]), THR2);
        total += S2 * (mA * accA + mB * accB);
    }

    // ---------------- level 3 : H4 (+identity padding) --------------------
    store_tile(T, d, n, h);
    load_B(T, B, n, h, /*transp=*/false);
    d = haar_mul(A4, B);
    store_tile(T, d, n, h);
    load_B(T, B, n, h, /*transp=*/true);
    d = haar_mul(A4, B);                    // (F*Z*F^T)^T, bands symmetric
    {
        float mA = (h == 0 && n >= 2 && n < 4) ? 1.0f : 0.0f;
        float mB = (h == 0 && n < 4) ? 1.0f : 0.0f;
        float accA = fminf(fabsf(d[0]), THR3) + fminf(fabsf(d[1]), THR3);
        float accB = fminf(fabsf(d[2]), THR3) + fminf(fabsf(d[3]), THR3);
        total += S3 * (mA * accA + mB * accB);
    }

    // ---------------- reduction: wave -> block -> d_ws slot ---------------
#pragma unroll
    for (int o = 16; o > 0; o >>= 1) total += __shfl_down(total, o, 32);
    if (lane == 0) wsum[wid] = total;
    __syncthreads();
    if (threadIdx.x == 0) {
        float s = 0.0f;
#pragma unroll
        for (int w = 0; w < 8; ++w) s += wsum[w];
        partials[blockIdx.x] = s;   // plain store: no same-address atomics
    }
}

extern "C" __global__ void __launch_bounds__(256)
haar_final_reduce(const float* __restrict__ partials, float* __restrict__ out, int nparts) {
    __shared__ float s[256];
    float acc = 0.0f;
    for (int i = threadIdx.x; i < nparts; i += 256) acc += partials[i];
    s[threadIdx.x] = acc;
    __syncthreads();
    for (int stride = 128; stride > 0; stride >>= 1) {
        if ((int)threadIdx.x < stride) s[threadIdx.x] += s[threadIdx.x + stride];
        __syncthreads();
    }
    if (threadIdx.x == 0) out[0] = s[0];
}

extern "C" void kernel_launch(void* const* d_in, const int* in_sizes, int n_in,
                              void* d_out, int out_size, void* d_ws, size_t ws_size,
                              hipStream_t stream) {
    const float* pred = (const float*)d_in[0];
    float* out      = (float*)d_out;
    float* partials = (float*)d_ws;          // needs ntiles/8 * 4 bytes = 128 KB

    const int ntiles  = in_sizes[0] / 256;   // 16x16 tiles: 262144
    const int nblocks = ntiles / 8;          // 8 waves (tiles) per 256-thr block

    haar_wavelet_loss<<<nblocks, 256, 0, stream>>>(pred, partials);
    haar_final_reduce<<<1, 256, 0, stream>>>(partials, out, nblocks);
}